// RN_18811956756917
// MI455X (gfx1250) — compile-verified
//
#include <hip/hip_runtime.h>

typedef float v2f __attribute__((ext_vector_type(2)));
typedef float v4f __attribute__((ext_vector_type(4)));
typedef float v8f __attribute__((ext_vector_type(8)));

#define B_   4
#define C_   32
#define HW   576        // N = H*W tokens per batch (24*24), 576 % 16 == 0
#define HID  128
#define OUT  32
#define ROWS (B_ * HW)  // 2304 total tokens

// ---------------------------------------------------------------------------
// Kernel 1: P[row, 0:256] = tokens @ [Wx | Wc]
//   tokens[row, c] = x[b, c, n]  (row = b*576 + n)
//   M = 2304, K = 32, N' = 256 (cols 0..127 = px, 128..255 = pc)
// One wave computes one 16x16 output tile via 8x v_wmma_f32_16x16x4_f32.
// ---------------------------------------------------------------------------
__global__ __launch_bounds__(256) void k1_proj(const float* __restrict__ x,
                                               const float* __restrict__ W1,
                                               float* __restrict__ P) {
  const int lane = threadIdx.x & 31;
  const int wave = threadIdx.x >> 5;
  const int tile = blockIdx.x * 8 + wave;   // 0..2303
  const int tm   = tile >> 4;               // 0..143 : 16-row tile of tokens
  const int tn   = tile & 15;               // 0..15  : 16-col tile of 256
  const int half = lane >> 4;               // 0/1 : k-pair select
  const int l15  = lane & 15;

  const int row0 = tm * 16;                 // tile never crosses a batch (576%16==0)
  const int b    = row0 / HW;
  const int n    = (row0 % HW) + l15;       // token position for this lane's A row
  const float* xb = x + b * (C_ * HW) + n;  // advance by c*HW per k

  // B operand: column of [Wx|Wc]; each tile lies fully in one half (tn<8 -> Wx)
  const int col = tn * 16 + l15;            // 0..255
  const float* Bbase = (col < HID) ? (W1 + col) : (W1 + C_ * HID + (col - HID));

  v8f acc = {};
#pragma unroll
  for (int k0 = 0; k0 < C_; k0 += 4) {
    const int c0 = k0 + (half << 1);        // lanes 0-15: k0,k0+1 ; 16-31: k0+2,k0+3
    v2f a, bf;
    a.x  = xb[c0 * HW];                     // lanes 0-15 contiguous in n -> 64B coalesced
    a.y  = xb[(c0 + 1) * HW];
    bf.x = Bbase[c0 * HID];                 // lanes contiguous in col -> 64B coalesced
    bf.y = Bbase[(c0 + 1) * HID];
    acc = __builtin_amdgcn_wmma_f32_16x16x4_f32(false, a, false, bf,
                                                (short)0, acc, false, false);
  }

  // C/D layout: VGPR r -> row (r + half*8), lane l15 -> col
  float* Pst = P + (row0 + half * 8) * 256 + tn * 16 + l15;
#pragma unroll
  for (int r = 0; r < 8; ++r)
    Pst[r * 256] = acc[r];
}

// ---------------------------------------------------------------------------
// Kernel 2: s[b,j,:] = sum_i relu(px[b,i,:] + pc[b,j,:] + b1)
// Elementwise-reduce: one wave per j, each lane owns 4 hid channels (float4).
// px panel for a batch (288 KB) is L2-resident and re-read by 576 waves (72 blocks).
// ---------------------------------------------------------------------------
__global__ __launch_bounds__(256) void k2_relusum(const float* __restrict__ P,
                                                  const float* __restrict__ b1,
                                                  float* __restrict__ S) {
  const int lane = threadIdx.x & 31;
  const int wave = threadIdx.x >> 5;
  const int row  = blockIdx.x * 8 + wave;   // 0..2303 : (b, j)
  const int b    = row / HW;
  const int ch   = lane * 4;                // 128 channels across 32 lanes

  v4f v = *(const v4f*)(P + row * 256 + HID + ch);   // pc[b,j,ch..]
  const v4f bias = *(const v4f*)(b1 + ch);
  v += bias;

  const float* px = P + (b * HW) * 256 + ch;
  v4f acc = {};
#pragma unroll 4
  for (int i = 0; i < HW; ++i) {
    v4f p = *(const v4f*)(px + i * 256);    // 512B/wave, fully coalesced
    p += v;
    acc.x += fmaxf(p.x, 0.f);
    acc.y += fmaxf(p.y, 0.f);
    acc.z += fmaxf(p.z, 0.f);
    acc.w += fmaxf(p.w, 0.f);
  }
  *(v4f*)(S + row * HID + ch) = acc;
}

// ---------------------------------------------------------------------------
// Kernel 3: rel = s @ W2 + 576*b2 ; out[b,o,h,w] = rel[b, n, o]
// M = 2304, K = 128, N' = 32 -> 288 tiles, 32 WMMAs each.
// Tile staged through LDS so output stores are contiguous along n.
// ---------------------------------------------------------------------------
__global__ __launch_bounds__(256) void k3_out(const float* __restrict__ S,
                                              const float* __restrict__ W2,
                                              const float* __restrict__ b2,
                                              float* __restrict__ out) {
  __shared__ float lds[8][16][17];          // +1 pad: conflict-free transpose
  const int lane = threadIdx.x & 31;
  const int wave = threadIdx.x >> 5;
  const int tile = blockIdx.x * 8 + wave;   // 0..287
  const int tm   = tile >> 1;               // 0..143
  const int tn   = tile & 1;                // 0..1
  const int half = lane >> 4;
  const int l15  = lane & 15;

  const float* Arow = S  + (tm * 16 + l15) * HID;  // A: row m = l15
  const float* Bcol = W2 + tn * 16 + l15;          // B: col n = l15

  v8f acc = {};
#pragma unroll
  for (int k0 = 0; k0 < HID; k0 += 4) {
    const int k = k0 + (half << 1);
    v2f a, bf;
    a.x  = Arow[k];
    a.y  = Arow[k + 1];
    bf.x = Bcol[k * OUT];
    bf.y = Bcol[(k + 1) * OUT];
    acc = __builtin_amdgcn_wmma_f32_16x16x4_f32(false, a, false, bf,
                                                (short)0, acc, false, false);
  }

  // Stage rel tile (m = token, c = out-channel) into LDS
#pragma unroll
  for (int r = 0; r < 8; ++r)
    lds[wave][r + half * 8][l15] = acc[r];
  __syncthreads();                          // no divergence: grid covers tiles exactly

  // Transposed store: 16 lanes sweep contiguous n for a fixed o -> 64B coalesced
  const int row0 = tm * 16;
  const int b    = row0 / HW;
  const int n0   = row0 % HW;
#pragma unroll
  for (int it = 0; it < 8; ++it) {
    const int c = half + it * 2;            // 0..15 within tile
    const int o = tn * 16 + c;
    const float val = lds[wave][l15][c] + 576.0f * b2[o];
    out[b * (OUT * HW) + o * HW + n0 + l15] = val;
  }
}

// ---------------------------------------------------------------------------
extern "C" void kernel_launch(void* const* d_in, const int* in_sizes, int n_in,
                              void* d_out, int out_size, void* d_ws, size_t ws_size,
                              hipStream_t stream) {
  const float* x  = (const float*)d_in[0];
  const float* W1 = (const float*)d_in[1];
  const float* b1 = (const float*)d_in[2];
  const float* W2 = (const float*)d_in[3];
  const float* b2 = (const float*)d_in[4];
  float* out = (float*)d_out;

  float* P = (float*)d_ws;                  // 2304*256 floats = 2.36 MB
  float* S = P + (size_t)ROWS * 256;        // 2304*128 floats = 1.18 MB

  // 2304 tiles / 8 waves per block
  k1_proj<<<ROWS * 16 / 8 / 16, 256, 0, stream>>>(x, W1, P);     // 288 blocks
  k2_relusum<<<ROWS / 8, 256, 0, stream>>>(P, b1, S);            // 288 blocks
  k3_out<<<(ROWS / 16) * 2 / 8, 256, 0, stream>>>(S, W2, b2, out); // 36 blocks
}